// EfficientGlobalPointer_48550310313986
// MI455X (gfx1250) — compile-verified
//
#include <hip/hip_runtime.h>

typedef float v2f __attribute__((ext_vector_type(2)));
typedef float v8f __attribute__((ext_vector_type(8)));

#define B_   16
#define L_   512
#define H_   768
#define D2_  128      // 2*HEAD_SIZE
#define HS_  64       // HEAD_SIZE
#define NH_  12       // HEADS
#define INF_ 10000.0f

// ---------------------------------------------------------------------------
// K1: X[8192,128] = inputs[8192,768] @ W1[768,128] + b1
// One wave per 16-row tile; computes all 8 column tiles so A is read once.
// A frag layout: lane%16 = M, k = vgpr + 2*(lane/16).
// B frag layout: lane%16 = N, k = vgpr + 2*(lane/16).
// D layout:      lane%16 = N, M = vgpr + 8*(lane/16).
// ---------------------------------------------------------------------------
__global__ __launch_bounds__(32)
void egp_gemm1(const float* __restrict__ inp, const float* __restrict__ W1,
               const float* __restrict__ b1, float* __restrict__ X) {
  const int lane = threadIdx.x;
  const int half = lane >> 4;
  const int l16  = lane & 15;
  const int rt   = blockIdx.x;              // row tile 0..511
  const float* arow = inp + (size_t)(rt * 16 + l16) * H_;

  v8f zero = {};
  v8f acc[8];
#pragma unroll
  for (int c = 0; c < 8; ++c) acc[c] = zero;

  for (int k0 = 0; k0 < H_; k0 += 4) {
    const int kk = k0 + 2 * half;           // k index for vgpr 0 of this lane
    v2f a = *(const v2f*)(arow + kk);
#pragma unroll
    for (int c = 0; c < 8; ++c) {
      v2f bf;
      bf.x = W1[(size_t)kk * D2_ + c * 16 + l16];
      bf.y = W1[(size_t)(kk + 1) * D2_ + c * 16 + l16];
      acc[c] = __builtin_amdgcn_wmma_f32_16x16x4_f32(
          false, a, false, bf, (short)0, acc[c], false, false);
    }
  }
#pragma unroll
  for (int c = 0; c < 8; ++c) {
    const float bias = b1[c * 16 + l16];
#pragma unroll
    for (int v = 0; v < 8; ++v) {
      const int m = rt * 16 + v + 8 * half;
      X[(size_t)m * D2_ + c * 16 + l16] = acc[c][v] + bias;
    }
  }
}

// ---------------------------------------------------------------------------
// K2a: RoPE. qw[j]=x[2j], kw[j]=x[2j+1]; rope pairs (qw[2i],qw[2i+1])=(x[4i],x[4i+2]).
// One thread per (token, i), i = 0..31 frequency index.
// ---------------------------------------------------------------------------
__global__ __launch_bounds__(256)
void egp_rope(const float* __restrict__ X, float* __restrict__ qr,
              float* __restrict__ kr) {
  const int tid = blockIdx.x * 256 + threadIdx.x;
  const int tok = tid >> 5;                 // 0..8191
  const int i   = tid & 31;
  const int l   = tok & (L_ - 1);           // position
  float4 x4 = *(const float4*)(X + (size_t)tok * D2_ + 4 * i);
  const float invf = powf(10000.0f, -(float)i * (1.0f / 32.0f));
  const float fr = (float)l * invf;
  const float s = sinf(fr), c = cosf(fr);
  float2 qo, ko;
  qo.x = x4.x * c - x4.z * s;  qo.y = x4.x * s + x4.z * c;
  ko.x = x4.y * c - x4.w * s;  ko.y = x4.y * s + x4.w * c;
  *(float2*)(qr + (size_t)tok * HS_ + 2 * i) = qo;
  *(float2*)(kr + (size_t)tok * HS_ + 2 * i) = ko;
}

// ---------------------------------------------------------------------------
// K2b: biasT[b,24,L] = (X @ W2 + b2) / 2, stored (channel-major, n fastest).
// ---------------------------------------------------------------------------
__global__ __launch_bounds__(256)
void egp_bias(const float* __restrict__ X, const float* __restrict__ W2,
              const float* __restrict__ b2, float* __restrict__ biasT) {
  const int idx = blockIdx.x * 256 + threadIdx.x;  // b*24*512 total
  const int l = idx & (L_ - 1);
  const int j = (idx >> 9) % 24;
  const int b = idx / (24 * L_);
  const float* xp = X + ((size_t)b * L_ + l) * D2_;
  float sum = 0.0f;
#pragma unroll 4
  for (int k = 0; k < D2_; ++k) sum += xp[k] * W2[k * 24 + j];
  biasT[idx] = 0.5f * (sum + b2[j]);
}

// ---------------------------------------------------------------------------
// K3: fused qk WMMA + bias broadcast + masks. One wave per (b, 16x16 tile).
// The qk tile lives in registers and is reused for all 12 heads, so the only
// large memory stream is the 201 MB of output writes.
// ---------------------------------------------------------------------------
__global__ __launch_bounds__(32)
void egp_fused(const float* __restrict__ qr, const float* __restrict__ kr,
               const float* __restrict__ biasT, const int* __restrict__ am,
               float* __restrict__ out) {
  const int lane = threadIdx.x;
  const int half = lane >> 4;
  const int l16  = lane & 15;
  const int n0 = blockIdx.x * 16;
  const int m0 = blockIdx.y * 16;
  const int b  = blockIdx.z;
  const float* qb = qr + (size_t)b * L_ * HS_;
  const float* kb = kr + (size_t)b * L_ * HS_;
  const int n = n0 + l16;

  v8f acc = {};
  for (int k0 = 0; k0 < HS_; k0 += 4) {
    v2f a  = *(const v2f*)(qb + (size_t)(m0 + l16) * HS_ + k0 + 2 * half);
    v2f bf = *(const v2f*)(kb + (size_t)n * HS_ + k0 + 2 * half);
    acc = __builtin_amdgcn_wmma_f32_16x16x4_f32(
        false, a, false, bf, (short)0, acc, false, false);
  }

  const int amn = am[b * L_ + n];
  int   mrow[8];
  float maskv[8];
#pragma unroll
  for (int v = 0; v < 8; ++v) {
    const int m = m0 + v + 8 * half;
    mrow[v] = m;
    float msk = 0.0f;
    if (m > n) msk -= INF_;                       // tril(k=-1) causal mask
    if (!(amn && am[b * L_ + m])) msk -= INF_;    // attention mask
    maskv[v] = msk;
  }

  const float scale = 0.125f;                     // 1/sqrt(64)
#pragma unroll
  for (int h = 0; h < NH_; ++h) {
    const float be = biasT[((size_t)b * 24 + 2 * h) * L_ + n];
    const float* bop = biasT + ((size_t)b * 24 + 2 * h + 1) * L_;
    float* op = out + ((size_t)(b * NH_ + h)) * L_ * L_;
#pragma unroll
    for (int v = 0; v < 8; ++v) {
      op[(size_t)mrow[v] * L_ + n] = acc[v] * scale + be + bop[mrow[v]] + maskv[v];
    }
  }
}

// ---------------------------------------------------------------------------
extern "C" void kernel_launch(void* const* d_in, const int* in_sizes, int n_in,
                              void* d_out, int out_size, void* d_ws, size_t ws_size,
                              hipStream_t stream) {
  const float* inp = (const float*)d_in[0];   // (16,512,768) f32
  const int*   am  = (const int*)d_in[1];     // (16,512)     i32
  const float* W1  = (const float*)d_in[2];   // (768,128)    f32
  const float* b1  = (const float*)d_in[3];   // (128,)       f32
  const float* W2  = (const float*)d_in[4];   // (128,24)     f32
  const float* b2  = (const float*)d_in[5];   // (24,)        f32
  float* out = (float*)d_out;                 // (16,12,512,512) f32

  float* X     = (float*)d_ws;                        // 8192*128
  float* qr    = X + (size_t)B_ * L_ * D2_;           // 8192*64
  float* kr    = qr + (size_t)B_ * L_ * HS_;          // 8192*64
  float* biasT = kr + (size_t)B_ * L_ * HS_;          // 16*24*512

  egp_gemm1<<<dim3(B_ * L_ / 16), dim3(32), 0, stream>>>(inp, W1, b1, X);
  egp_rope <<<dim3(B_ * L_ * 32 / 256), dim3(256), 0, stream>>>(X, qr, kr);
  egp_bias <<<dim3(B_ * 24 * L_ / 256), dim3(256), 0, stream>>>(X, W2, b2, biasT);
  egp_fused<<<dim3(L_ / 16, L_ / 16, B_), dim3(32), 0, stream>>>(qr, kr, biasT, am, out);
}